// Op_4389456577013
// MI455X (gfx1250) — compile-verified
//
#include <hip/hip_runtime.h>
#include <hip/hip_bf16.h>

// ---------------------------------------------------------------------------
// Problem constants (reference: B=64, TAPE=262144, OUT=65536, FANIN=16)
// ---------------------------------------------------------------------------
constexpr int kB     = 64;
constexpr int kTAPE  = 262144;
constexpr int kOUT   = 65536;
constexpr int kFANIN = 16;

// Block geometry: 256 threads = 64 outputs x 4 batch-slices (16 batches each).
// -> 1024 blocks, 8192 wave32s: plenty of MLP for the random gathers.
constexpr int kBlock         = 256;
constexpr int kOutsPerBlock  = 64;
constexpr int kSlices        = kBlock / kOutsPerBlock;   // 4
constexpr int kBatchPerSlice = kB / kSlices;             // 16

typedef float f4  __attribute__((ext_vector_type(4)));
typedef int   v4i __attribute__((ext_vector_type(4)));

#if defined(__gfx1250__) && __has_builtin(__builtin_amdgcn_global_load_async_to_lds_b128)
#define USE_ASYNC_LDS 1
#else
#define USE_ASYNC_LDS 0
#endif

// ---------------------------------------------------------------------------
// Kernel: gathered weighted sum + per-node activation + scatter.
//   thread t: olane = t & 63 (output within block), slice = t >> 6 (batch 1/4)
//   Waves 0..1 (slice==0, wave-uniform) stage the block's 64 rows of
//   indices/weights into LDS via CDNA5 async global->LDS DMA.
// ---------------------------------------------------------------------------
__global__ __launch_bounds__(kBlock) void nca_gather_kernel(
    const float* __restrict__ tape, const float* __restrict__ weights,
    const float* __restrict__ bias, const int* __restrict__ in_idx,
    const int* __restrict__ out_idx, const int* __restrict__ act_ids,
    float* __restrict__ out) {
  const int t     = threadIdx.x;
  const int olane = t & (kOutsPerBlock - 1);
  const int slice = t >> 6;
  const int o     = blockIdx.x * kOutsPerBlock + olane;

  int   li[kFANIN];
  float lw[kFANIN];

#if USE_ASYNC_LDS
  __shared__ int   s_idx[kOutsPerBlock * kFANIN];
  __shared__ float s_w[kOutsPerBlock * kFANIN];
  if (slice == 0) {  // wave-uniform: waves 0 and 1 of the block
    typedef __attribute__((address_space(1))) v4i* gvp;
    typedef __attribute__((address_space(3))) v4i* lvp;
    gvp gi = (gvp)(in_idx + (size_t)o * kFANIN);     // 64 B row = 4 x b128
    lvp si = (lvp)(&s_idx[olane * kFANIN]);
    gvp gw = (gvp)(weights + (size_t)o * kFANIN);
    lvp sw = (lvp)(&s_w[olane * kFANIN]);
    __builtin_amdgcn_global_load_async_to_lds_b128(gi + 0, si + 0, 0, 0);
    __builtin_amdgcn_global_load_async_to_lds_b128(gi + 1, si + 1, 0, 0);
    __builtin_amdgcn_global_load_async_to_lds_b128(gi + 2, si + 2, 0, 0);
    __builtin_amdgcn_global_load_async_to_lds_b128(gi + 3, si + 3, 0, 0);
    __builtin_amdgcn_global_load_async_to_lds_b128(gw + 0, sw + 0, 0, 0);
    __builtin_amdgcn_global_load_async_to_lds_b128(gw + 1, sw + 1, 0, 0);
    __builtin_amdgcn_global_load_async_to_lds_b128(gw + 2, sw + 2, 0, 0);
    __builtin_amdgcn_global_load_async_to_lds_b128(gw + 3, sw + 3, 0, 0);
#if __has_builtin(__builtin_amdgcn_s_wait_asynccnt)
    __builtin_amdgcn_s_wait_asynccnt(0);
#else
    asm volatile("s_wait_asynccnt 0" ::: "memory");
#endif
  }
  __syncthreads();
#pragma unroll
  for (int f = 0; f < kFANIN; ++f) {
    li[f] = s_idx[olane * kFANIN + f];
    lw[f] = s_w[olane * kFANIN + f];
  }
#else
#pragma unroll
  for (int f = 0; f < kFANIN; ++f) {
    li[f] = in_idx[(size_t)o * kFANIN + f];
    lw[f] = weights[(size_t)o * kFANIN + f];
  }
#endif

  const float vb = bias[o];
  const int   a  = act_ids[o];
  const int   oo = out_idx[o];

  const int b0 = slice * kBatchPerSlice;
#pragma unroll 2
  for (int bb = 0; bb < kBatchPerSlice; ++bb) {
    const int b = b0 + bb;
    const float* __restrict__ row = tape + (size_t)b * kTAPE;
    // 16 independent gathers; 4-way accumulator tree for ILP
    float a0 = 0.f, a1 = 0.f, a2 = 0.f, a3 = 0.f;
#pragma unroll
    for (int f = 0; f < kFANIN; f += 4) {
      a0 = fmaf(row[li[f + 0]], lw[f + 0], a0);
      a1 = fmaf(row[li[f + 1]], lw[f + 1], a1);
      a2 = fmaf(row[li[f + 2]], lw[f + 2], a2);
      a3 = fmaf(row[li[f + 3]], lw[f + 3], a3);
    }
    const float acc = vb + ((a0 + a1) + (a2 + a3));

    float r;
    if (a == 0) {
      r = fmaxf(acc, 0.0f);                          // relu
    } else if (a == 1) {
      r = 1.0f / (1.0f + __expf(-acc));              // sigmoid (v_exp_f32)
    } else {
      r = 1.0f - 2.0f / (__expf(2.0f * acc) + 1.0f); // tanh, saturates to +-1
    }
    out[(size_t)b * kTAPE + oo] = r;                 // coalesced per wave
  }
}

// ---------------------------------------------------------------------------
// Kernel: copy tape -> out (67 MB each way).  Non-temporal stores: the
// destination copy is never re-read by this dispatch.
// ---------------------------------------------------------------------------
__global__ __launch_bounds__(kBlock) void tape_copy_kernel(
    const float* __restrict__ src, float* __restrict__ dst, int n) {
  const int n4 = n >> 2;
  const f4* __restrict__ s4 = (const f4*)src;
  f4* __restrict__ d4 = (f4*)dst;
  int i = blockIdx.x * blockDim.x + threadIdx.x;
  const int stride = gridDim.x * blockDim.x;
  for (; i < n4; i += stride) {
    f4 v = s4[i];
    __builtin_nontemporal_store(v, &d4[i]);
  }
  // scalar tail (robustness; n is a multiple of 4 for this problem)
  const int tail = n4 << 2;
  if (blockIdx.x == 0) {
    for (int j = tail + threadIdx.x; j < n; j += blockDim.x) dst[j] = src[j];
  }
}

// ---------------------------------------------------------------------------
// Launch
// ---------------------------------------------------------------------------
extern "C" void kernel_launch(void* const* d_in, const int* in_sizes, int n_in,
                              void* d_out, int out_size, void* d_ws, size_t ws_size,
                              hipStream_t stream) {
  const float* tape    = (const float*)d_in[0];  // [B, TAPE] f32
  const float* weights = (const float*)d_in[1];  // [OUT, FANIN] f32
  const float* bias    = (const float*)d_in[2];  // [OUT] f32
  const int*   in_idx  = (const int*)d_in[3];    // [OUT, FANIN] int
  const int*   out_idx = (const int*)d_in[4];    // [OUT] int
  const int*   act_ids = (const int*)d_in[5];    // [OUT] int
  float*       out     = (float*)d_out;          // [B, TAPE] f32
  (void)in_sizes; (void)n_in; (void)d_ws; (void)ws_size;

  // 1) copy tape into output (scatter targets get overwritten below)
  tape_copy_kernel<<<4096, kBlock, 0, stream>>>(tape, out, out_size);

  // 2) gather + weighted-sum + activation + scatter
  nca_gather_kernel<<<kOUT / kOutsPerBlock, kBlock, 0, stream>>>(
      tape, weights, bias, in_idx, out_idx, act_ids, out);
}